// TransitionUp_661424963760
// MI455X (gfx1250) — compile-verified
//
#include <hip/hip_runtime.h>

typedef float v2f __attribute__((ext_vector_type(2)));
typedef float v8f __attribute__((ext_vector_type(8)));

#define BN_EPS    1e-5f
#define CLAMP_MIN 1e-16f
#define B_        8
#define N_        2048
#define M_        8192
#define CIN_      256
#define COUT_     128

// ---------------------------------------------------------------------------
// Fused 1x1-Conv (GEMM) + BatchNorm(eval) + ReLU using native fp32 WMMA
// (V_WMMA_F32_16X16X4_F32, wave32). One block (8 waves) computes a 16-row
// stripe across all 128 output channels; wave w owns col-tile w.
// The 16 x KD A-stripe is contiguous in X -> staged into LDS with the CDNA5
// async copy engine (GLOBAL_LOAD_ASYNC_TO_LDS_B128, tracked on ASYNCcnt:
// no VGPR round-trip). Padded LDS stride KD+4 floats -> fragment reads hit
// banks {4i,4i+1}/{4i+2,4i+3}: conflict-free, full 64-bank use.
// B fragments (weights, <=128KB) read from global: L2-resident on 192MB L2.
// Fragment layouts per CDNA5 ISA 7.12.2:
//   A 16x4  : lane<16 -> K={0,1}, lane>=16 -> K={2,3}; row M = lane%16
//   B 4x16  : lanes0-15 K={0,1}, lanes16-31 K={2,3};  col N = lane%16
//   C 16x16 : acc[v] -> M = v + 8*(lane>=16), N = lane%16
// ---------------------------------------------------------------------------
template <int KD>
__global__ __launch_bounds__(256) void mlp_wmma_kernel(
    const float* __restrict__ X, const float* __restrict__ W,
    const float* __restrict__ g, const float* __restrict__ beta,
    const float* __restrict__ mu, const float* __restrict__ var,
    float* __restrict__ Y)
{
    constexpr int LDK = KD + 4;                 // padded LDS row stride
    __shared__ float As[16 * LDK];

    const int tm   = blockIdx.x;                // 16-row stripe index
    const int lane = threadIdx.x & 31;
    const int tn   = threadIdx.x >> 5;          // wave id == column tile
    const int half = lane >> 4;
    const int l16  = lane & 15;

    // Async stage of the contiguous 16 x KD A-stripe into padded LDS.
    // Each lane issues an independent 16B global->LDS copy; ASYNCcnt tracks
    // completion, so each wave drains with s_wait_asynccnt 0 and the
    // workgroup barrier then guarantees the whole stripe is resident.
    const float4* __restrict__ Xs = (const float4*)(X + (size_t)tm * 16 * KD);
#pragma unroll
    for (int i = threadIdx.x; i < 16 * KD / 4; i += 256) {
        const int row = i / (KD / 4);
        const int c4  = i % (KD / 4);
        // LDS byte offset = low 32 bits of the generic shared-space address
        const unsigned lds_addr =
            (unsigned)(uintptr_t)(&As[row * LDK + c4 * 4]);
        const float4* gsrc = &Xs[i];
        asm volatile("global_load_async_to_lds_b128 %0, %1, off"
                     :: "v"(lds_addr), "v"(gsrc) : "memory");
    }
    asm volatile("s_wait_asynccnt 0x0" ::: "memory");
    __syncthreads();

    const int ocol = tn * 16 + l16;             // output channel (B column)
    const float* __restrict__ Bp = W + (size_t)ocol * KD;  // B = W^T

    v8f acc = {};
#pragma unroll 8
    for (int k = 0; k < KD; k += 4) {
        const v2f a = *(const v2f*)(&As[l16 * LDK + k + half * 2]);
        const v2f b = *(const v2f*)(Bp + k + half * 2);
        acc = __builtin_amdgcn_wmma_f32_16x16x4_f32(
                  /*neg_a=*/false, a, /*neg_b=*/false, b,
                  /*c_mod=*/(short)0, acc, /*reuse_a=*/false, /*reuse_b=*/false);
    }

    // Fused BN(eval) + ReLU epilogue; all 8 acc elems share channel `ocol`.
    const float s   = g[ocol] * rsqrtf(var[ocol] + BN_EPS);
    const float off = beta[ocol] - mu[ocol] * s;
#pragma unroll
    for (int v = 0; v < 8; ++v) {
        const int r = tm * 16 + v + half * 8;
        Y[(size_t)r * COUT_ + ocol] = fmaxf(fmaf(acc[v], s, off), 0.0f);
    }
}

// ---------------------------------------------------------------------------
// Brute-force 3-NN: one thread per query. Source cloud staged as float4 in
// LDS (32KB of the 320KB WGP pool) -> one ds_load_b128 broadcast per point
// (all lanes read the same address). Staging stays synchronous: it must
// repack xyz -> xyzw, which the async copy engine cannot do, and it is a
// one-off 24KB cost vs the 2048-iteration scan it accelerates.
// Branchy top-3 insert: once the running top-3 tightens, `d < b2` fails in
// all 32 lanes and the wave skips the shift logic (wave32-friendly).
// ---------------------------------------------------------------------------
__global__ __launch_bounds__(256) void knn_kernel(
    const float* __restrict__ p, const float* __restrict__ p_old,
    int* __restrict__ idx_out, float* __restrict__ w_out)
{
    __shared__ float4 sp[N_];
    const int b = blockIdx.y;
    const int m = blockIdx.x * 256 + threadIdx.x;

    const float* __restrict__ pb = p + (size_t)b * N_ * 3;
    for (int j = threadIdx.x; j < N_; j += 256) {
        sp[j] = make_float4(pb[j * 3 + 0], pb[j * 3 + 1], pb[j * 3 + 2], 0.0f);
    }
    __syncthreads();

    const float* __restrict__ q = p_old + ((size_t)b * M_ + m) * 3;
    const float qx = q[0], qy = q[1], qz = q[2];

    float b0 = 3.4e38f, b1 = 3.4e38f, b2 = 3.4e38f;
    int   i0 = 0,       i1 = 0,       i2 = 0;

#pragma unroll 4
    for (int j = 0; j < N_; ++j) {
        const float4 s = sp[j];
        const float dx = qx - s.x;
        const float dy = qy - s.y;
        const float dz = qz - s.z;
        const float d  = fmaf(dx, dx, fmaf(dy, dy, dz * dz));
        if (d < b2) {
            if (d < b0)      { b2 = b1; i2 = i1; b1 = b0; i1 = i0; b0 = d; i0 = j; }
            else if (d < b1) { b2 = b1; i2 = i1; b1 = d;  i1 = j; }
            else             { b2 = d;  i2 = j; }
        }
    }

    const size_t o = ((size_t)b * M_ + m) * 3;
    idx_out[o + 0] = i0;  idx_out[o + 1] = i1;  idx_out[o + 2] = i2;
    w_out[o + 0] = 1.0f / fmaxf(b0, CLAMP_MIN);
    w_out[o + 1] = 1.0f / fmaxf(b1, CLAMP_MIN);
    w_out[o + 2] = 1.0f / fmaxf(b2, CLAMP_MIN);
}

// ---------------------------------------------------------------------------
// Gather 3 neighbor feature rows (float4-vectorized: 32 lanes x 16B = one
// dense 512B row per neighbor), inverse-d2 weighted average, accumulate into
// y (which already holds x_lat). One wave per query, 4 queries per block.
// ---------------------------------------------------------------------------
__global__ __launch_bounds__(128) void gather_add_kernel(
    const float* __restrict__ xup, const int* __restrict__ idx,
    const float* __restrict__ w, float* __restrict__ y)
{
    const int q    = blockIdx.x * 4 + (threadIdx.x >> 5);  // query id
    const int lane = threadIdx.x & 31;                     // float4 slot (128/4)
    const int b    = q / M_;

    const size_t o = (size_t)q * 3;
    const int   i0 = idx[o], i1 = idx[o + 1], i2 = idx[o + 2];
    const float w0 = w[o],   w1 = w[o + 1],   w2 = w[o + 2];
    const float invden = 1.0f / (w0 + w1 + w2);

    const float4* __restrict__ xb =
        (const float4*)(xup + (size_t)b * N_ * COUT_);
    const float4 a0 = xb[(size_t)i0 * (COUT_ / 4) + lane];
    const float4 a1 = xb[(size_t)i1 * (COUT_ / 4) + lane];
    const float4 a2 = xb[(size_t)i2 * (COUT_ / 4) + lane];

    float4* __restrict__ y4 = (float4*)y + (size_t)q * (COUT_ / 4) + lane;
    float4 r = *y4;
    r.x += (w0 * a0.x + w1 * a1.x + w2 * a2.x) * invden;
    r.y += (w0 * a0.y + w1 * a1.y + w2 * a2.y) * invden;
    r.z += (w0 * a0.z + w1 * a1.z + w2 * a2.z) * invden;
    r.w += (w0 * a0.w + w1 * a1.w + w2 * a2.w) * invden;
    *y4 = r;
}

extern "C" void kernel_launch(void* const* d_in, const int* in_sizes, int n_in,
                              void* d_out, int out_size, void* d_ws, size_t ws_size,
                              hipStream_t stream) {
    (void)in_sizes; (void)n_in; (void)out_size; (void)ws_size;

    const float* x     = (const float*)d_in[0];   // (B,N,Cin)
    const float* p     = (const float*)d_in[1];   // (B,N,3)
    const float* x_old = (const float*)d_in[2];   // (B,M,Cout)
    const float* p_old = (const float*)d_in[3];   // (B,M,3)
    const float* W_up  = (const float*)d_in[4];   // (Cout,Cin)
    const float* g1    = (const float*)d_in[5];
    const float* bb1   = (const float*)d_in[6];
    const float* m1    = (const float*)d_in[7];
    const float* v1    = (const float*)d_in[8];
    const float* W_lat = (const float*)d_in[9];   // (Cout,Cout)
    const float* g2    = (const float*)d_in[10];
    const float* bb2   = (const float*)d_in[11];
    const float* m2    = (const float*)d_in[12];
    const float* v2    = (const float*)d_in[13];

    float* y    = (float*)d_out;                          // (B,M,Cout)
    float* pout = y + (size_t)B_ * M_ * COUT_;            // (B,M,3) passthrough

    // Workspace: x_up (B*N*Cout f32) | idx (B*M*3 i32) | w (B*M*3 f32)
    float* xup  = (float*)d_ws;
    int*   idx  = (int*)(xup + (size_t)B_ * N_ * COUT_);
    float* wbuf = (float*)(idx + (size_t)B_ * M_ * 3);

    // 1) x_up = relu(bn(x @ W_up^T))   rows=16384, K=256, cols=128
    mlp_wmma_kernel<CIN_><<<B_ * N_ / 16, 256, 0, stream>>>(
        x, W_up, g1, bb1, m1, v1, xup);

    // 2) x_lat = relu(bn(x_old @ W_lat^T)) written directly into y
    mlp_wmma_kernel<COUT_><<<B_ * M_ / 16, 256, 0, stream>>>(
        x_old, W_lat, g2, bb2, m2, v2, y);

    // 3) 3-NN indices + inverse-d2 weights
    {
        dim3 grid(M_ / 256, B_);
        knn_kernel<<<grid, 256, 0, stream>>>(p, p_old, idx, wbuf);
    }

    // 4) y += knn_interpolate(x_up)
    gather_add_kernel<<<B_ * M_ / 4, 128, 0, stream>>>(xup, idx, wbuf, y);

    // 5) second output: p_old passthrough
    hipMemcpyAsync(pout, p_old, (size_t)B_ * M_ * 3 * sizeof(float),
                   hipMemcpyDeviceToDevice, stream);
}